// GraphSAGELayer_1554778161866
// MI455X (gfx1250) — compile-verified
//
#include <hip/hip_runtime.h>
#include <hip/hip_bf16.h>

typedef float v2f __attribute__((ext_vector_type(2)));
typedef float v8f __attribute__((ext_vector_type(8)));

#define D_FEAT 128   // D_IN == D_OUT == 128

// -------------------------------------------------------------------------
// Stage 1: h = x @ W^T + b   via V_WMMA_F32_16X16X4_F32
// Block = 256 threads = 8 waves. Each wave computes a 16-row x 128-col strip.
// W (128x128, 64 KB) staged into LDS transposed: sWt[k*128 + n] = W[n*128 + k]
// so B-fragment reads hit consecutive banks across lanes.
// -------------------------------------------------------------------------
__global__ void __launch_bounds__(256) sage_gemm_wmma(
    const float* __restrict__ x, const float* __restrict__ W,
    const float* __restrict__ bias, float* __restrict__ h, int n_nodes) {
  __shared__ float sWt[D_FEAT * D_FEAT];  // 64 KB (well under 320 KB/WGP)

  const int tid = threadIdx.x;
  // Cooperative transposed load of W into LDS (coalesced global reads).
  for (int i = tid; i < D_FEAT * D_FEAT; i += 256) {
    const int n = i >> 7;       // row of W  (output feature)
    const int k = i & 127;      // col of W  (input feature)
    sWt[k * D_FEAT + n] = W[i];
  }
  __syncthreads();

  const int wave = tid >> 5;
  const int lane = tid & 31;
  const int rowBase = (blockIdx.x * 8 + wave) * 16;
  if (rowBase >= n_nodes) return;  // wave-uniform exit (EXEC stays all-1s)

  const int ml = lane & 15;        // m (for A) / n (for B,D) within 16-tile
  const int kh = (lane >> 4) * 2;  // lane-half K offset (K=0,1 vs K=2,3)

  int arow = rowBase + ml;
  if (arow >= n_nodes) arow = n_nodes - 1;  // clamp (keeps EXEC full)
  const float* __restrict__ xrow = x + (size_t)arow * D_FEAT;

  const v8f vzero = {0.f, 0.f, 0.f, 0.f, 0.f, 0.f, 0.f, 0.f};
  v8f acc[8];
#pragma unroll
  for (int t = 0; t < 8; ++t) acc[t] = vzero;

  for (int kb = 0; kb < D_FEAT; kb += 4) {
    // A fragment: lane holds x[row=ml][kb+kh .. kb+kh+1]  (8B-aligned b64 load)
    v2f a = *(const v2f*)(xrow + kb + kh);
#pragma unroll
    for (int t = 0; t < 8; ++t) {
      const int n = t * 16 + ml;
      // B fragment: B[k][n] = W[n][k] -> transposed LDS, conflict-free
      v2f bf;
      bf.x = sWt[(kb + kh) * D_FEAT + n];
      bf.y = sWt[(kb + kh + 1) * D_FEAT + n];
      acc[t] = __builtin_amdgcn_wmma_f32_16x16x4_f32(
          /*neg_a=*/false, a, /*neg_b=*/false, bf,
          /*c_mod=*/(short)0, acc[t], /*reuse_a=*/false, /*reuse_b=*/false);
    }
  }

  // D layout: acc[t][r] = D[m = r + 8*(lane>>4)][n = t*16 + ml]
  const int mh = (lane >> 4) * 8;
#pragma unroll
  for (int t = 0; t < 8; ++t) {
    const int n = t * 16 + ml;
    const float bv = bias[n];
#pragma unroll
    for (int r = 0; r < 8; ++r) {
      const int m = rowBase + mh + r;
      if (m < n_nodes) h[(size_t)m * D_FEAT + n] = acc[t][r] + bv;
    }
  }
}

// -------------------------------------------------------------------------
// Stage 2: edge scatter. One wave per edge; lane owns 4 features.
// agg[row] += w * h[col]  via global_atomic_add_f32; deg[row] += w (lane 0).
// -------------------------------------------------------------------------
__global__ void __launch_bounds__(256) sage_scatter(
    const long long* __restrict__ ei, const float* __restrict__ ew,
    const float* __restrict__ h, float* __restrict__ agg,
    float* __restrict__ deg, int n_edges) {
  const int lane = threadIdx.x & 31;
  const int e = blockIdx.x * 8 + (threadIdx.x >> 5);
  if (e >= n_edges) return;

  const size_t dst = (size_t)ei[e];            // row
  const size_t src = (size_t)ei[n_edges + e];  // col
  const float w = ew[e];

  const float4 hv = *(const float4*)(h + src * D_FEAT + lane * 4);
  float* ap = agg + dst * D_FEAT + lane * 4;
  atomicAdd(ap + 0, hv.x * w);
  atomicAdd(ap + 1, hv.y * w);
  atomicAdd(ap + 2, hv.z * w);
  atomicAdd(ap + 3, hv.w * w);
  if (lane == 0) atomicAdd(deg + dst, w);
}

// -------------------------------------------------------------------------
// Stage 3: out = normalize(agg * deg_inv). One wave per node; wave32
// butterfly reduction for the row L2 norm.
// -------------------------------------------------------------------------
__global__ void __launch_bounds__(256) sage_normalize(
    float* __restrict__ out, const float* __restrict__ deg, int n_nodes) {
  const int lane = threadIdx.x & 31;
  const int node = blockIdx.x * 8 + (threadIdx.x >> 5);
  if (node >= n_nodes) return;

  const float d = deg[node];
  const float dinv = d > 0.f ? 1.f / d : 0.f;

  float* p = out + (size_t)node * D_FEAT + lane * 4;
  float4 v = *(const float4*)p;
  v.x *= dinv; v.y *= dinv; v.z *= dinv; v.w *= dinv;

  float ss = v.x * v.x + v.y * v.y + v.z * v.z + v.w * v.w;
#pragma unroll
  for (int off = 16; off > 0; off >>= 1) ss += __shfl_xor(ss, off, 32);

  const float nrm = sqrtf(ss);
  const float scale = 1.f / fmaxf(nrm, 1e-12f);
  v.x *= scale; v.y *= scale; v.z *= scale; v.w *= scale;
  *(float4*)p = v;
}

// -------------------------------------------------------------------------
extern "C" void kernel_launch(void* const* d_in, const int* in_sizes, int n_in,
                              void* d_out, int out_size, void* d_ws,
                              size_t ws_size, hipStream_t stream) {
  const float*     x    = (const float*)d_in[0];
  const long long* ei   = (const long long*)d_in[1];  // int64 per reference
  const float*     ew   = (const float*)d_in[2];
  const float*     W    = (const float*)d_in[3];
  const float*     bias = (const float*)d_in[4];
  float*           out  = (float*)d_out;

  const int n_nodes = in_sizes[0] / D_FEAT;  // 20000
  const int n_edges = in_sizes[2];           // 640000

  float* h   = (float*)d_ws;                  // [n_nodes, 128]
  float* deg = h + (size_t)n_nodes * D_FEAT;  // [n_nodes]

  // Zero accumulation targets every launch (harness poisons buffers).
  hipMemsetAsync(out, 0, (size_t)n_nodes * D_FEAT * sizeof(float), stream);
  hipMemsetAsync(deg, 0, (size_t)n_nodes * sizeof(float), stream);

  const dim3 blk(256);
  sage_gemm_wmma<<<dim3((n_nodes + 127) / 128), blk, 0, stream>>>(
      x, W, bias, h, n_nodes);
  sage_scatter<<<dim3((n_edges + 7) / 8), blk, 0, stream>>>(
      ei, ew, h, out, deg, n_edges);
  sage_normalize<<<dim3((n_nodes + 7) / 8), blk, 0, stream>>>(
      out, deg, n_nodes);
}